// GCNDecoder_45827301048642
// MI455X (gfx1250) — compile-verified
//
#include <hip/hip_runtime.h>
#include <hip/hip_bf16.h>
#include <cstddef>

#define CDIV(a,b) (((a)+(b)-1)/(b))

typedef __attribute__((ext_vector_type(2))) float v2f;
typedef __attribute__((ext_vector_type(8))) float v8f;

// ------------------------------------------------------------------
// helpers
// ------------------------------------------------------------------
__device__ __forceinline__ void atomAddF(float* p, float v) {
  unsafeAtomicAdd(p, v);   // global_atomic_add_f32 on gfx1250
}

// ------------------------------------------------------------------
// degree prep: deg starts at 1 (self loop), += 1 per edge col, rsqrt
// ------------------------------------------------------------------
__global__ void k_fill(float* __restrict__ p, float v, int n) {
  int i = blockIdx.x * blockDim.x + threadIdx.x;
  if (i < n) p[i] = v;
}

__global__ void k_deg_accum(const int* __restrict__ col, float* __restrict__ deg, int E) {
  int e = blockIdx.x * blockDim.x + threadIdx.x;
  if (e < E) atomAddF(&deg[col[e]], 1.0f);
}

__global__ void k_rsqrt_inplace(float* __restrict__ d, int n) {
  int i = blockIdx.x * blockDim.x + threadIdx.x;
  if (i < n) d[i] = rsqrtf(d[i]);
}

// ------------------------------------------------------------------
// GEMM: C[N x 64] = A[N x K] @ W^T + bias, W is [64 x K] row-major.
// Templated on K so the K-loop fully unrolls into immediate-offset
// global_load_b64s. SPLIT=true: A = [A0 | A1] split at column 64
// (both sub-matrices row stride 64 floats) for the merge layer.
//
// One wave computes a full 16x64 output block (4 n-tiles): the A
// operand (v2f per lane per K-step) is register-reused across the 4
// V_WMMA_F32_16X16X4_F32 issues of each K-step.
//
// A 16x4 f32 layout: lanes 0-15 -> M=lane, {k0,k0+1}; lanes 16-31 -> {k0+2,k0+3}.
// B 4x16 f32 layout: mirrored; B[k][n] = W[n0+n][k0+k].
// C/D: vgpr r: lanes 0-15 -> M=m0+r, lanes 16-31 -> M=m0+8+r, N = n0 + lane%16.
// ------------------------------------------------------------------
template<int K, bool SPLIT>
__global__ void k_gemm_wmma_f32(const float* __restrict__ A0,
                                const float* __restrict__ A1,
                                const float* __restrict__ W,
                                const float* __restrict__ bias,
                                float* __restrict__ C,
                                int Nrows) {
  const int lane = threadIdx.x & 31;
  const int wave = threadIdx.x >> 5;
  const int mt = blockIdx.x * 8 + wave;
  const int m0 = mt * 16;
  if (m0 >= Nrows) return;
  const int lm = lane & 15;        // 0..15
  const int hi = lane >> 4;        // 0 / 1

  const float* a0p = A0 + (size_t)(m0 + lm) * 64 + 2 * hi;
  const float* a1p = SPLIT ? (A1 + (size_t)(m0 + lm) * 64 + 2 * hi) : a0p;
  const float* wp[4];
#pragma unroll
  for (int nt = 0; nt < 4; ++nt)
    wp[nt] = W + (size_t)(nt * 16 + lm) * K + 2 * hi;

  v8f acc[4] = {{}, {}, {}, {}};
#pragma unroll
  for (int k0 = 0; k0 < K; k0 += 4) {
    v2f a;
    if (SPLIT && k0 >= 64) a = *(const v2f*)(a1p + (k0 - 64));
    else                   a = *(const v2f*)(a0p + k0);
    v2f b0 = *(const v2f*)(wp[0] + k0);
    v2f b1 = *(const v2f*)(wp[1] + k0);
    v2f b2 = *(const v2f*)(wp[2] + k0);
    v2f b3 = *(const v2f*)(wp[3] + k0);
    acc[0] = __builtin_amdgcn_wmma_f32_16x16x4_f32(false, a, false, b0, (short)0, acc[0], false, false);
    acc[1] = __builtin_amdgcn_wmma_f32_16x16x4_f32(false, a, false, b1, (short)0, acc[1], false, false);
    acc[2] = __builtin_amdgcn_wmma_f32_16x16x4_f32(false, a, false, b2, (short)0, acc[2], false, false);
    acc[3] = __builtin_amdgcn_wmma_f32_16x16x4_f32(false, a, false, b3, (short)0, acc[3], false, false);
  }

  const int rbase = m0 + 8 * hi;
#pragma unroll
  for (int nt = 0; nt < 4; ++nt) {
    const float bi = bias[nt * 16 + lm];
    float* cp = C + (size_t)rbase * 64 + nt * 16 + lm;
#pragma unroll
    for (int r = 0; r < 8; ++r) {
      cp[(size_t)r * 64] = acc[nt][r] + bi;
    }
  }
}

// ------------------------------------------------------------------
// GCN aggregation: out[i] = dis[i]^2 * h[i]  (self loop init)
// then out[col] += dis[row]*dis[col] * h[row] per edge (atomic scatter)
// ------------------------------------------------------------------
__global__ void k_selfinit(const float* __restrict__ h, const float* __restrict__ dis,
                           float* __restrict__ out, int n64) {
  int i = blockIdx.x * blockDim.x + threadIdx.x;
  if (i >= n64) return;
  float d = dis[i >> 6];
  out[i] = d * d * h[i];
}

__global__ void k_scatter(const float* __restrict__ h,
                          const int* __restrict__ row,
                          const int* __restrict__ col,
                          const float* __restrict__ dis,
                          float* __restrict__ out, int E) {
  long long t = (long long)blockIdx.x * blockDim.x + threadIdx.x;
  int e = (int)(t >> 4);
  if (e >= E) return;
  int c = ((int)t & 15) * 4;
  int r = row[e], cc = col[e];
  float norm = dis[r] * dis[cc];
  const float4 v = *(const float4*)(h + (size_t)r * 64 + c);
  float* dst = out + (size_t)cc * 64 + c;
  atomAddF(dst + 0, norm * v.x);
  atomAddF(dst + 1, norm * v.y);
  atomAddF(dst + 2, norm * v.z);
  atomAddF(dst + 3, norm * v.w);
}

// ------------------------------------------------------------------
// BatchNorm (training-mode batch stats) + activation
// ------------------------------------------------------------------
__global__ void k_bn_stats(const float* __restrict__ x, float* __restrict__ stats,
                           int Nrows) {
  __shared__ float ssum[4][64];
  __shared__ float ssq[4][64];
  const int c = threadIdx.x & 63;
  const int g = threadIdx.x >> 6;          // 0..3
  float s = 0.f, q = 0.f;
  for (int r = blockIdx.x * 4 + g; r < Nrows; r += gridDim.x * 4) {
    float v = x[(size_t)r * 64 + c];
    s += v;
    q += v * v;
  }
  ssum[g][c] = s;
  ssq[g][c]  = q;
  __syncthreads();
  if (threadIdx.x < 64) {
    int cc = threadIdx.x;
    float S = ssum[0][cc] + ssum[1][cc] + ssum[2][cc] + ssum[3][cc];
    float Q = ssq[0][cc] + ssq[1][cc] + ssq[2][cc] + ssq[3][cc];
    atomAddF(&stats[cc], S);
    atomAddF(&stats[64 + cc], Q);
  }
}

// leaky=0 -> ReLU ; leaky=1 -> LeakyReLU(0.1)
__global__ void k_bn_apply(const float* __restrict__ x, const float* __restrict__ stats,
                           const float* __restrict__ gamma, const float* __restrict__ beta,
                           float* __restrict__ y, int n64, float invN, int leaky) {
  int i = blockIdx.x * blockDim.x + threadIdx.x;
  if (i >= n64) return;
  int c = i & 63;
  float mu = stats[c] * invN;
  float var = stats[64 + c] * invN - mu * mu;
  float sc = gamma[c] * rsqrtf(var + 1e-5f);
  float sh = beta[c] - mu * sc;
  float v = fmaf(x[i], sc, sh);
  y[i] = leaky ? (v >= 0.f ? v : 0.1f * v) : fmaxf(v, 0.f);
}

// ------------------------------------------------------------------
// PowerMean pooling: clip -> pow(p) -> mean over 10000 nodes -> clip -> pow(1/p)
// one block per (batch, channel)
// ------------------------------------------------------------------
__global__ void k_pool_pm(const float* __restrict__ g, const float* __restrict__ p_ptr,
                          float* __restrict__ pooled, int nodesPerGraph) {
  const int b = blockIdx.x >> 6;
  const int c = blockIdx.x & 63;
  const float p = p_ptr[0];
  float s = 0.f;
  for (int r = threadIdx.x; r < nodesPerGraph; r += blockDim.x) {
    float v = g[((size_t)b * nodesPerGraph + r) * 64 + c];
    v = fminf(fmaxf(v, 1e-7f), 100.0f);
    s += __powf(v, p);
  }
  __shared__ float red[256];
  red[threadIdx.x] = s;
  __syncthreads();
  for (int w = 128; w > 0; w >>= 1) {
    if (threadIdx.x < w) red[threadIdx.x] += red[threadIdx.x + w];
    __syncthreads();
  }
  if (threadIdx.x == 0) {
    float m = red[0] / (float)nodesPerGraph;
    m = fminf(fmaxf(m, 1e-7f), 100.0f);
    pooled[b * 64 + c] = __powf(m, 1.0f / p);
  }
}

// ------------------------------------------------------------------
// Final head: logits[8x10] = pooled[8x64] @ Wp^T + bp ; ypred = argmax
// out layout: 80 logits then (if room) 8 predictions (as float values)
// ------------------------------------------------------------------
__global__ void k_head(const float* __restrict__ pooled, const float* __restrict__ Wp,
                       const float* __restrict__ bp, float* __restrict__ out,
                       int out_size) {
  __shared__ float logits[8][10];
  int t = threadIdx.x;
  if (t < 80) {
    int b = t / 10, k = t % 10;
    float acc = bp[k];
#pragma unroll
    for (int c = 0; c < 64; ++c) acc = fmaf(pooled[b * 64 + c], Wp[k * 64 + c], acc);
    logits[b][k] = acc;
    out[t] = acc;
  }
  __syncthreads();
  if (t < 8 && out_size >= 88) {
    int best = 0;
    float bv = logits[t][0];
#pragma unroll
    for (int k = 1; k < 10; ++k)
      if (logits[t][k] > bv) { bv = logits[t][k]; best = k; }
    out[80 + t] = (float)best;
  }
}

// ------------------------------------------------------------------
// host side
// ------------------------------------------------------------------
extern "C" void kernel_launch(void* const* d_in, const int* in_sizes, int n_in,
                              void* d_out, int out_size, void* d_ws, size_t ws_size,
                              hipStream_t stream) {
  const float* x   = (const float*)d_in[0];
  const int* ei    = (const int*)d_in[1];
  const int* ea    = (const int*)d_in[2];
  const float* Wt  = (const float*)d_in[3];   const float* bt  = (const float*)d_in[4];
  const float* Wm  = (const float*)d_in[5];   const float* bm  = (const float*)d_in[6];
  const float* Wi[3]  = { (const float*)d_in[7],  (const float*)d_in[9],  (const float*)d_in[11] };
  const float* bi[3]  = { (const float*)d_in[8],  (const float*)d_in[10], (const float*)d_in[12] };
  const float* Wg[3]  = { (const float*)d_in[13], (const float*)d_in[15], (const float*)d_in[17] };
  const float* bg[3]  = { (const float*)d_in[14], (const float*)d_in[16], (const float*)d_in[18] };
  const float* gi[3]  = { (const float*)d_in[19], (const float*)d_in[21], (const float*)d_in[23] };
  const float* bbi[3] = { (const float*)d_in[20], (const float*)d_in[22], (const float*)d_in[24] };
  const float* gg[3]  = { (const float*)d_in[25], (const float*)d_in[27], (const float*)d_in[29] };
  const float* bbg[3] = { (const float*)d_in[26], (const float*)d_in[28], (const float*)d_in[30] };
  const float* pexp   = (const float*)d_in[31];
  const float* Wp     = (const float*)d_in[32]; const float* bp = (const float*)d_in[33];

  const int N     = in_sizes[0] / 64;     // 80000
  const int E_int = in_sizes[1] / 2;      // 640000
  const int E_all = in_sizes[2] / 2;      // 1280000
  const int n64   = N * 64;
  const float invN = 1.0f / (float)N;
  const int nodesPerGraph = 10000;

  const int* row_i = ei;            const int* col_i = ei + E_int;
  const int* row_a = ea;            const int* col_a = ea + E_all;

  // workspace layout (3 ping-pong feature buffers + norm vectors + stats)
  float* ws = (float*)d_ws;
  size_t NB = (size_t)n64;
  float* bufA  = ws;
  float* bufB  = bufA + NB;
  float* bufC  = bufB + NB;
  float* dis_i = bufC + NB;
  float* dis_a = dis_i + N;
  float* stats = dis_a + N;         // 128 floats
  float* pooled = stats + 128;      // 512 floats

  const int T = 256;
  const int gemmBlocks = CDIV(N / 16, 8);   // 1 wave per 16x64 output block

  // ---- degree / normalization ----
  k_fill<<<CDIV(N, T), T, 0, stream>>>(dis_i, 1.0f, N);
  k_fill<<<CDIV(N, T), T, 0, stream>>>(dis_a, 1.0f, N);
  k_deg_accum<<<CDIV(E_int, T), T, 0, stream>>>(col_i, dis_i, E_int);
  k_deg_accum<<<CDIV(E_all, T), T, 0, stream>>>(col_a, dis_a, E_all);
  k_rsqrt_inplace<<<CDIV(N, T), T, 0, stream>>>(dis_i, N);
  k_rsqrt_inplace<<<CDIV(N, T), T, 0, stream>>>(dis_a, N);

  // ---- internal_transform: A = x @ Wt^T + bt ----
  k_gemm_wmma_f32<64, false><<<gemmBlocks, T, 0, stream>>>(x, nullptr, Wt, bt, bufA, N);

  // ---- 3 internal GCN layers (ReLU) : h lives in bufA ----
  for (int l = 0; l < 3; ++l) {
    k_gemm_wmma_f32<64, false><<<gemmBlocks, T, 0, stream>>>(bufA, nullptr, Wi[l], bi[l], bufB, N);
    k_selfinit<<<CDIV(n64, T), T, 0, stream>>>(bufB, dis_i, bufC, n64);
    k_scatter<<<CDIV(E_int * 16, T), T, 0, stream>>>(bufB, row_i, col_i, dis_i, bufC, E_int);
    k_fill<<<1, 128, 0, stream>>>(stats, 0.0f, 128);
    k_bn_stats<<<512, T, 0, stream>>>(bufC, stats, N);
    k_bn_apply<<<CDIV(n64, T), T, 0, stream>>>(bufC, stats, gi[l], bbi[l], bufA, n64, invN, 0);
  }

  // ---- merge: g = [x | h] @ Wm^T + bm  (K = 128, split A) -> bufB ----
  k_gemm_wmma_f32<128, true><<<gemmBlocks, T, 0, stream>>>(x, bufA, Wm, bm, bufB, N);

  // ---- 3 global GCN layers (LeakyReLU 0.1) : g lives in bufB ----
  for (int l = 0; l < 3; ++l) {
    k_gemm_wmma_f32<64, false><<<gemmBlocks, T, 0, stream>>>(bufB, nullptr, Wg[l], bg[l], bufC, N);
    k_selfinit<<<CDIV(n64, T), T, 0, stream>>>(bufC, dis_a, bufA, n64);
    k_scatter<<<CDIV(E_all * 16, T), T, 0, stream>>>(bufC, row_a, col_a, dis_a, bufA, E_all);
    k_fill<<<1, 128, 0, stream>>>(stats, 0.0f, 128);
    k_bn_stats<<<512, T, 0, stream>>>(bufA, stats, N);
    k_bn_apply<<<CDIV(n64, T), T, 0, stream>>>(bufA, stats, gg[l], bbg[l], bufB, n64, invN, 1);
  }

  // ---- PowerMean pooling + head ----
  k_pool_pm<<<8 * 64, T, 0, stream>>>(bufB, pexp, pooled, nodesPerGraph);
  k_head<<<1, 128, 0, stream>>>(pooled, Wp, bp, (float*)d_out, out_size);

  (void)n_in; (void)ws_size;
}